// ArcTransformer_1546188226567
// MI455X (gfx1250) — compile-verified
//
#include <hip/hip_runtime.h>
#include <hip/hip_bf16.h>
#include <cstdint>
#include <cstddef>

// ---------------------------------------------------------------------------
// Model constants (from reference)
// ---------------------------------------------------------------------------
#define BB      2
#define TWOP    8
#define PP      4            // TWO_P / 2
#define PTR     3            // P - 1
#define HH      26
#define WW      26
#define LL      676          // H*W
#define BE      16           // B*TWO_P
#define BP      8            // B*P
#define IND     11
#define DD      256
#define FFD     1024
#define OUTD    11
#define NH      8
#define DH      32
#define NPFc    64
#define NTFc    128
#define RDIMc   155
#define RPAD    160          // RDIM padded to multiple of 4 (and 32)
#define EXTP    1480         // 2L + NTF
#define NEGV    (-1e9f)
#define SCALE_A 0.17677669529663687f  // 1/sqrt(32)

typedef float v2f __attribute__((ext_vector_type(2)));
typedef float v8f __attribute__((ext_vector_type(8)));

__device__ __forceinline__ v8f wmma4(v2f a, v2f b, v8f c) {
  // D = A(16x4,f32) * B(4x16,f32) + C(16x16,f32)
  return __builtin_amdgcn_wmma_f32_16x16x4_f32(false, a, false, b, (short)0, c,
                                               false, false);
}

// ---------------------------------------------------------------------------
// WMMA GEMM: C[M,N] = act(A[M,K] @ W[K,N] + bias[N])
// one wave per 16x64 tile; requires M % 16 == 0, N % 64 == 0, K % 4 == 0.
// No lane-dependent guards -> tight branch-free inner loop.
// ---------------------------------------------------------------------------
__global__ __launch_bounds__(32) void k_gemm(const float* __restrict__ A,
                                             const float* __restrict__ W,
                                             const float* __restrict__ bias,
                                             float* __restrict__ C,
                                             int M, int K, int N, int relu) {
  const int lane = threadIdx.x;
  const int l15  = lane & 15;
  const int koff = (lane >> 4) << 1;      // K sub-offset per lane half
  const int m0   = blockIdx.y << 4;
  const int n0   = blockIdx.x << 6;
  const int mrow = m0 + l15;
  const float* __restrict__ arow = A + (size_t)mrow * K + koff;
  const float* __restrict__ wcol = W + (size_t)koff * N + n0 + l15;

  v8f acc0 = {}; v8f acc1 = {}; v8f acc2 = {}; v8f acc3 = {};

  #pragma unroll 2
  for (int k = 0; k < K; k += 4) {
    const v2f a = *(const v2f*)(arow);            // A[m][kb], A[m][kb+1]
    const float* __restrict__ w0 = wcol;          // W[kb][n..]
    const float* __restrict__ w1 = wcol + N;      // W[kb+1][n..]
    if (k + 16 < K) __builtin_prefetch(w0 + (size_t)16 * N, 0, 1);
    v2f b0, b1, b2, b3;
    b0.x = w0[0];  b0.y = w1[0];
    b1.x = w0[16]; b1.y = w1[16];
    b2.x = w0[32]; b2.y = w1[32];
    b3.x = w0[48]; b3.y = w1[48];
    acc0 = wmma4(a, b0, acc0);
    acc1 = wmma4(a, b1, acc1);
    acc2 = wmma4(a, b2, acc2);
    acc3 = wmma4(a, b3, acc3);
    arow += 4;
    wcol += (size_t)4 * N;
  }

  const float bi0 = bias[n0 + l15];
  const float bi1 = bias[n0 + l15 + 16];
  const float bi2 = bias[n0 + l15 + 32];
  const float bi3 = bias[n0 + l15 + 48];
  const int rbase = m0 + ((lane >> 4) << 3);
  #pragma unroll
  for (int v = 0; v < 8; ++v) {
    const int row = rbase + v;
    float* __restrict__ crow = C + (size_t)row * N + n0 + l15;
    float o0 = acc0[v] + bi0;
    float o1 = acc1[v] + bi1;
    float o2 = acc2[v] + bi2;
    float o3 = acc3[v] + bi3;
    if (relu) { o0 = fmaxf(o0, 0.f); o1 = fmaxf(o1, 0.f);
                o2 = fmaxf(o2, 0.f); o3 = fmaxf(o3, 0.f); }
    crow[0] = o0; crow[16] = o1; crow[32] = o2; crow[48] = o3;
  }
}

// ---------------------------------------------------------------------------
// Fused attention: one wave per (q-tile of 16, head, batch).
// Q,K,V seq-major: [(seq*Bq + b)*256 + h*32 + d]. R batch-major:
// [((b*Lk + k)*256) + h*32 + d]. mask: [b*Lk + k] (broadcast over q).
// Scores staged in LDS [16][Lkp]; softmax in LDS; P@V via WMMA from LDS.
// ---------------------------------------------------------------------------
__global__ __launch_bounds__(32) void k_attn(const float* __restrict__ Q,
                                             const float* __restrict__ Kt,
                                             const float* __restrict__ V,
                                             const float* __restrict__ R,
                                             const float* __restrict__ maskb,
                                             float* __restrict__ O,
                                             int Lq, int Lk, int Bq) {
  extern __shared__ float sm[];               // 16 * Lkp floats
  const int lane  = threadIdx.x;
  const int l15   = lane & 15;
  const int lhalf = lane >> 4;
  const int koff  = lhalf << 1;
  const int q0    = blockIdx.x << 4;
  const int h     = blockIdx.y;
  const int b     = blockIdx.z;
  const int Lkp   = (Lk + 15) & ~15;

  // ---- pass A: S = (Q K^T + Q R^T) * scale + mask  ->  LDS -----------------
  int mq = q0 + l15; if (mq >= Lq) mq = Lq - 1;
  const float* __restrict__ qrow = Q + ((size_t)mq * Bq + b) * DD + h * DH + koff;

  for (int kt = 0; kt < Lkp; kt += 16) {
    const int nk  = kt + l15;
    const int nkc = (nk < Lk) ? nk : (Lk - 1);
    const float* __restrict__ krow = Kt + ((size_t)nkc * Bq + b) * DD + h * DH + koff;
    const float* __restrict__ rrow =
        R ? (R + ((size_t)b * Lk + nkc) * DD + h * DH + koff) : nullptr;
    v8f acc = {};
    #pragma unroll
    for (int dd = 0; dd < DH; dd += 4) {
      const v2f a  = *(const v2f*)(qrow + dd);
      const v2f bk = *(const v2f*)(krow + dd);
      acc = wmma4(a, bk, acc);
      if (R) { const v2f br = *(const v2f*)(rrow + dd);
               acc = wmma4(a, br, acc); }
    }
    float mval = 0.f;
    if (maskb) mval = maskb[(size_t)b * Lk + nkc];
    #pragma unroll
    for (int v = 0; v < 8; ++v) {
      float s = acc[v] * SCALE_A + mval;
      if (nk >= Lk) s = -1e30f;
      sm[(size_t)(v + lhalf * 8) * Lkp + nk] = s;
    }
  }
  __syncthreads();

  // ---- softmax (one lane per q row) ---------------------------------------
  if (lane < 16) {
    float* row = sm + (size_t)lane * Lkp;
    float mx = -1e30f;
    for (int c = 0; c < Lk; ++c) mx = fmaxf(mx, row[c]);
    float ssum = 0.f;
    for (int c = 0; c < Lk; ++c) { float e = __expf(row[c] - mx); row[c] = e; ssum += e; }
    const float inv = 1.f / ssum;
    for (int c = 0; c < Lk; ++c) row[c] *= inv;
    for (int c = Lk; c < Lkp; ++c) row[c] = 0.f;
  }
  __syncthreads();

  // ---- pass B: O = P @ V ---------------------------------------------------
  v8f o0 = {}; v8f o1 = {};
  const float* __restrict__ prow = sm + (size_t)l15 * Lkp + koff;
  for (int kt = 0; kt < Lkp; kt += 16) {
    #pragma unroll
    for (int ks = 0; ks < 16; ks += 4) {
      const int kb = kt + ks + koff;
      const v2f a = *(const v2f*)(prow + kt + ks);
      const int k0c = (kb     < Lk) ? kb     : (Lk - 1);
      const int k1c = (kb + 1 < Lk) ? kb + 1 : (Lk - 1);
      const float* __restrict__ v0 = V + ((size_t)k0c * Bq + b) * DD + h * DH;
      const float* __restrict__ v1 = V + ((size_t)k1c * Bq + b) * DD + h * DH;
      v2f b0; b0.x = v0[l15];      b0.y = v1[l15];
      v2f b1; b1.x = v0[16 + l15]; b1.y = v1[16 + l15];
      o0 = wmma4(a, b0, o0);
      o1 = wmma4(a, b1, o1);
    }
  }
  #pragma unroll
  for (int v = 0; v < 8; ++v) {
    const int qr = q0 + v + lhalf * 8;
    if (qr < Lq) {
      float* __restrict__ orow = O + ((size_t)qr * Bq + b) * DD + h * DH;
      orow[l15]      = o0[v];
      orow[16 + l15] = o1[v];
    }
  }
}

// ---------------------------------------------------------------------------
// Residual + LayerNorm over D=256 (one wave per row, wave32 shfl reductions)
// ---------------------------------------------------------------------------
__global__ __launch_bounds__(32) void k_add_ln(const float* __restrict__ X,
                                               const float* __restrict__ Yr,
                                               const float* __restrict__ gam,
                                               const float* __restrict__ bet,
                                               float* __restrict__ Oo) {
  const int row  = blockIdx.x;
  const int lane = threadIdx.x;
  const float* __restrict__ x = X  + (size_t)row * DD;
  const float* __restrict__ y = Yr + (size_t)row * DD;
  float v[8];
  float s = 0.f;
  #pragma unroll
  for (int i = 0; i < 8; ++i) { const int c = lane + 32 * i; v[i] = x[c] + y[c]; s += v[i]; }
  #pragma unroll
  for (int o = 16; o > 0; o >>= 1) s += __shfl_xor(s, o, 32);
  const float mu = s * (1.f / 256.f);
  float vs = 0.f;
  #pragma unroll
  for (int i = 0; i < 8; ++i) { const float d = v[i] - mu; vs += d * d; }
  #pragma unroll
  for (int o = 16; o > 0; o >>= 1) vs += __shfl_xor(vs, o, 32);
  const float inv = rsqrtf(vs * (1.f / 256.f) + 1e-5f);
  #pragma unroll
  for (int i = 0; i < 8; ++i) {
    const int c = lane + 32 * i;
    Oo[(size_t)row * DD + c] = (v[i] - mu) * inv * gam[c] + bet[c];
  }
}

// ---------------------------------------------------------------------------
// Elementwise / setup kernels
// ---------------------------------------------------------------------------
__global__ void k_gmask(const float* __restrict__ grids,
                        float* __restrict__ gmask, float* __restrict__ maskA) {
  const int t = blockIdx.x * blockDim.x + threadIdx.x;
  if (t >= BE * LL) return;
  const int be = t / LL, l = t % LL;
  float s = 0.f;
  #pragma unroll
  for (int i = 0; i < IND; ++i) s += grids[((size_t)be * LL + l) * IND + i];
  const bool on = s > 0.f;
  gmask[t] = on ? 1.f : 0.f;
  maskA[t] = on ? 0.f : NEGV;
}

__global__ void k_maskE(const float* __restrict__ gmask, float* __restrict__ maskE) {
  const int t = blockIdx.x * blockDim.x + threadIdx.x;
  if (t >= BP * EXTP) return;
  const int bp = t / EXTP, kk = t % EXTP;
  float mv = 0.f;
  if (kk < 2 * LL) {
    const int b = bp >> 2, pp = bp & 3;
    const int e = pp * 2 + kk / LL, l = kk % LL;
    mv = (gmask[(size_t)(b * TWOP + e) * LL + l] == 0.f) ? NEGV : 0.f;
  }
  maskE[t] = mv;
}

// zero-pad prior (BE*L, 155) -> (BE*L, 160) so GEMMs never need K-guards
__global__ void k_pad_prior(const float* __restrict__ prior, float* __restrict__ pp) {
  const int t = blockIdx.x * blockDim.x + threadIdx.x;
  if (t >= BE * LL * RPAD) return;
  const int r = t / RPAD, c = t % RPAD;
  pp[t] = (c < RDIMc) ? prior[(size_t)r * RDIMc + c] : 0.f;
}

// zero-pad Wr (155, 256) -> (160, 256)
__global__ void k_pad_wr(const float* __restrict__ Wr, float* __restrict__ WrP) {
  const int t = blockIdx.x * blockDim.x + threadIdx.x;
  if (t >= RPAD * DD) return;
  const int k = t >> 8;
  WrP[t] = (k < RDIMc) ? Wr[t] : 0.f;
}

__global__ __launch_bounds__(32) void k_embed(const float* __restrict__ grids,
                                              const float* __restrict__ W_in,
                                              const float* __restrict__ b_in,
                                              const float* __restrict__ grid_pos,
                                              const float* __restrict__ tr_emb,
                                              const float* __restrict__ te_emb,
                                              const float* __restrict__ to_emb,
                                              const float* __restrict__ gmask,
                                              float* __restrict__ g) {
  const int row = blockIdx.x;                 // row = l*16 + be
  const int l = row >> 4, be = row & 15;
  const float gmv = gmask[(size_t)be * LL + l];
  const int w = l / HH, hh = l % HH;
  float gr[IND];
  #pragma unroll
  for (int i = 0; i < IND; ++i) gr[i] = grids[((size_t)be * LL + l) * IND + i];
  #pragma unroll
  for (int j = 0; j < 8; ++j) {
    const int d = threadIdx.x + 32 * j;
    float acc = b_in[d];
    #pragma unroll
    for (int i = 0; i < IND; ++i) acc += gr[i] * W_in[i * DD + d];
    const float gp = grid_pos[(size_t)(w * 30 + hh) * DD + d];
    float se = (l < PTR * 2) ? tr_emb[d] : te_emb[d];
    if (l == LL - 1) se += to_emb[d];
    g[(size_t)row * DD + d] = (acc + 2.f * gp + se) * gmv;
  }
}

__global__ void k_pf_init(const float* __restrict__ pair_pos,
                          const float* __restrict__ tr_emb,
                          const float* __restrict__ te_emb,
                          float* __restrict__ pf) {
  const int t = blockIdx.x * blockDim.x + threadIdx.x;
  if (t >= NPFc * BP * DD) return;
  const int d = t & 255, n = t >> 11;
  pf[t] = pair_pos[(size_t)n * DD + d] + ((n < PTR) ? tr_emb[d] : te_emb[d]);
}

__global__ void k_tf_init(const float* __restrict__ task_pos, float* __restrict__ tf) {
  const int t = blockIdx.x * blockDim.x + threadIdx.x;
  if (t >= NTFc * BB * DD) return;
  const int d = t & 255, n = t >> 9;
  tf[t] = task_pos[(size_t)n * DD + d];
}

__global__ void k_repeat2(const float* __restrict__ pf, float* __restrict__ extg) {
  const int t = blockIdx.x * blockDim.x + threadIdx.x;
  if (t >= NPFc * BE * DD) return;
  const int d = t & 255, be = (t >> 8) & 15, n = t >> 12;
  extg[t] = pf[((size_t)n * BP + (be >> 1)) * DD + d];
}

__global__ void k_scale_g(float* __restrict__ g, const float* __restrict__ gmask) {
  const int t = blockIdx.x * blockDim.x + threadIdx.x;
  if (t >= LL * BE * DD) return;
  const int be = (t >> 8) & 15, l = t >> 12;
  g[t] *= gmask[(size_t)be * LL + l];
}

__global__ void k_addio(float* __restrict__ g, const float* __restrict__ in_emb,
                        const float* __restrict__ out_emb) {
  const int t = blockIdx.x * blockDim.x + threadIdx.x;
  if (t >= LL * BE * DD) return;
  const int d = t & 255, be = (t >> 8) & 15;
  g[t] += (be & 1) ? out_emb[d] : in_emb[d];
}

__global__ void k_build_extp(const float* __restrict__ g, const float* __restrict__ tf,
                             const float* __restrict__ task_emb,
                             float* __restrict__ extp) {
  const int t = blockIdx.x * blockDim.x + threadIdx.x;
  if (t >= EXTP * BP * DD) return;
  const int d = t & 255, bp = (t >> 8) & 7, r = t >> 11;
  float val;
  if (r < 2 * LL) {
    const int c = r / LL, l = r % LL;
    val = g[((size_t)l * BE + (bp * 2 + c)) * DD + d];
  } else {
    const int n = r - 2 * LL;
    val = tf[((size_t)n * BB + (bp & 1)) * DD + d] + task_emb[d];
  }
  extp[t] = val;
}

__global__ void k_ext2(const float* __restrict__ pf, const float* __restrict__ pair_emb,
                       float* __restrict__ ext2) {
  const int t = blockIdx.x * blockDim.x + threadIdx.x;
  if (t >= PP * NPFc * BB * DD) return;
  const int d = t & 255, f = t >> 8;
  ext2[t] = pf[t] + pair_emb[(size_t)(f >> 7) * DD + d];
}

__global__ void k_out_proj(const float* __restrict__ g, const float* __restrict__ W_out,
                           const float* __restrict__ b_out, float* __restrict__ out) {
  const int t = blockIdx.x * blockDim.x + threadIdx.x;
  if (t >= LL * BE * OUTD) return;
  const int r = t / OUTD, j = t % OUTD;
  float acc = b_out[j];
  for (int k = 0; k < DD; ++k) acc += g[(size_t)r * DD + k] * W_out[k * OUTD + j];
  out[t] = acc;
}

// ---------------------------------------------------------------------------
// Host-side orchestration
// ---------------------------------------------------------------------------
struct AttnP { const float *Wq,*Wk,*Wv,*Wo,*bq,*bk,*bv,*bo; };
struct LayerP { AttnP self, ext;
                const float *W1,*b1,*W2,*b2,*g1,*be1,*g2,*be2,*g3,*be3; };

static inline void launch_gemm(hipStream_t s, const float* A, const float* W,
                               const float* bias, float* C, int M, int K, int N,
                               int relu) {
  dim3 grd((unsigned)(N / 64), (unsigned)(M / 16));
  k_gemm<<<grd, 32, 0, s>>>(A, W, bias, C, M, K, N, relu);
}

static inline void launch_attn(hipStream_t s, const float* Q, const float* K,
                               const float* V, const float* R, const float* mask,
                               float* O, int Lq, int Lk, int Bq) {
  const int Lkp = (Lk + 15) & ~15;
  dim3 grd((unsigned)((Lq + 15) / 16), NH, (unsigned)Bq);
  k_attn<<<grd, 32, (size_t)16 * Lkp * sizeof(float), s>>>(Q, K, V, R, mask, O,
                                                           Lq, Lk, Bq);
}

static inline void launch_ln(hipStream_t s, const float* X, const float* Y,
                             const float* g, const float* b, float* O, int T) {
  k_add_ln<<<(unsigned)T, 32, 0, s>>>(X, Y, g, b, O);
}

static void run_layer(hipStream_t s, const LayerP& L, float* tgt, int Lq, int Bq,
                      const float* ext, int Le, const float* Rrel,
                      const float* maskT, const float* maskE,
                      float* qb, float* kb, float* vb, float* ab,
                      float* ob, float* f1, float* f2) {
  const int T = Lq * Bq, Tk = Le * Bq;
  // self-attention
  launch_gemm(s, tgt, L.self.Wq, L.self.bq, qb, T, DD, DD, 0);
  launch_gemm(s, tgt, L.self.Wk, L.self.bk, kb, T, DD, DD, 0);
  launch_gemm(s, tgt, L.self.Wv, L.self.bv, vb, T, DD, DD, 0);
  launch_attn(s, qb, kb, vb, Rrel, maskT, ab, Lq, Lq, Bq);
  launch_gemm(s, ab, L.self.Wo, L.self.bo, ob, T, DD, DD, 0);
  launch_ln(s, tgt, ob, L.g1, L.be1, tgt, T);
  // external attention
  launch_gemm(s, tgt, L.ext.Wq, L.ext.bq, qb, T, DD, DD, 0);
  launch_gemm(s, ext, L.ext.Wk, L.ext.bk, kb, Tk, DD, DD, 0);
  launch_gemm(s, ext, L.ext.Wv, L.ext.bv, vb, Tk, DD, DD, 0);
  launch_attn(s, qb, kb, vb, nullptr, maskE, ab, Lq, Le, Bq);
  launch_gemm(s, ab, L.ext.Wo, L.ext.bo, ob, T, DD, DD, 0);
  launch_ln(s, tgt, ob, L.g2, L.be2, tgt, T);
  // FFN
  launch_gemm(s, tgt, L.W1, L.b1, f1, T, DD, FFD, 1);
  launch_gemm(s, f1, L.W2, L.b2, f2, T, FFD, DD, 0);
  launch_ln(s, tgt, f2, L.g3, L.be3, tgt, T);
}

extern "C" void kernel_launch(void* const* d_in, const int* in_sizes, int n_in,
                              void* d_out, int out_size, void* d_ws, size_t ws_size,
                              hipStream_t stream) {
  (void)in_sizes; (void)n_in; (void)out_size;
  auto P = [&](int i) { return (const float*)d_in[i]; };
  const float* grids = P(0);
  const float* prior = P(1);
  // params flattened in setup_inputs() insertion order (depth-first):
  const float* in_emb   = P(2);
  const float* out_emb  = P(3);
  const float* tr_emb   = P(4);
  const float* te_emb   = P(5);
  const float* to_emb   = P(6);
  const float* task_emb = P(7);
  const float* pair_emb = P(8);
  const float* grid_pos = P(9);
  const float* pair_pos = P(10);
  const float* task_pos = P(11);
  const float* W_in  = P(12);
  const float* b_in  = P(13);
  const float* W_out = P(14);
  const float* b_out = P(15);
  LayerP GL, PL, TL;
  GL.self = {P(16),P(17),P(18),P(19),P(20),P(21),P(22),P(23)};
  const float* Wr = P(24); const float* br = P(25);
  GL.ext  = {P(26),P(27),P(28),P(29),P(30),P(31),P(32),P(33)};
  GL.W1=P(34); GL.b1=P(35); GL.W2=P(36); GL.b2=P(37);
  GL.g1=P(38); GL.be1=P(39); GL.g2=P(40); GL.be2=P(41); GL.g3=P(42); GL.be3=P(43);
  PL.self = {P(44),P(45),P(46),P(47),P(48),P(49),P(50),P(51)};
  PL.ext  = {P(52),P(53),P(54),P(55),P(56),P(57),P(58),P(59)};
  PL.W1=P(60); PL.b1=P(61); PL.W2=P(62); PL.b2=P(63);
  PL.g1=P(64); PL.be1=P(65); PL.g2=P(66); PL.be2=P(67); PL.g3=P(68); PL.be3=P(69);
  TL.self = {P(70),P(71),P(72),P(73),P(74),P(75),P(76),P(77)};
  TL.ext  = {P(78),P(79),P(80),P(81),P(82),P(83),P(84),P(85)};
  TL.W1=P(86); TL.b1=P(87); TL.W2=P(88); TL.b2=P(89);
  TL.g1=P(90); TL.be1=P(91); TL.g2=P(92); TL.be2=P(93); TL.g3=P(94); TL.be3=P(95);

  // workspace arena (floats)
  float* ws = (float*)d_ws;
  size_t off = 0;
  auto alloc = [&](size_t n) { float* p = ws + off; off += n; return p; };
  float* gmask  = alloc((size_t)BE * LL);
  float* maskA  = alloc((size_t)BE * LL);
  float* maskE  = alloc((size_t)BP * EXTP);
  float* g      = alloc((size_t)LL * BE * DD);
  float* Rg     = alloc((size_t)BE * LL * DD);
  float* pf     = alloc((size_t)NPFc * BP * DD);
  float* tf     = alloc((size_t)NTFc * BB * DD);
  float* extg   = alloc((size_t)NPFc * BE * DD);
  float* extp   = alloc((size_t)EXTP * BP * DD);
  float* ext2   = alloc((size_t)PP * NPFc * BB * DD);
  float* priorP = alloc((size_t)BE * LL * RPAD);  // zero-padded prior
  float* WrP    = alloc((size_t)RPAD * DD);       // zero-padded Wr
  float* qb     = alloc((size_t)EXTP * BP * DD);  // max key rows = 11840
  float* kb     = alloc((size_t)EXTP * BP * DD);
  float* vb     = alloc((size_t)EXTP * BP * DD);
  float* ab     = alloc((size_t)EXTP * BP * DD);
  float* ob     = alloc((size_t)LL * BE * DD);
  float* f2b    = alloc((size_t)LL * BE * DD);
  float* f1b    = qb;  // FFN hidden aliases q/k/v/a (not live during FFN)
  if (ws_size < off * sizeof(float)) return;

  const int TPB = 256;
  auto blocks = [&](int n) { return (unsigned)((n + TPB - 1) / TPB); };

  // setup
  k_gmask<<<blocks(BE * LL), TPB, 0, stream>>>(grids, gmask, maskA);
  k_maskE<<<blocks(BP * EXTP), TPB, 0, stream>>>(gmask, maskE);
  k_embed<<<(unsigned)(LL * BE), 32, 0, stream>>>(grids, W_in, b_in, grid_pos,
                                                  tr_emb, te_emb, to_emb, gmask, g);
  k_pf_init<<<blocks(NPFc * BP * DD), TPB, 0, stream>>>(pair_pos, tr_emb, te_emb, pf);
  k_tf_init<<<blocks(NTFc * BB * DD), TPB, 0, stream>>>(task_pos, tf);
  // relative prior projection (computed once; K padded 155 -> 160)
  k_pad_prior<<<blocks(BE * LL * RPAD), TPB, 0, stream>>>(prior, priorP);
  k_pad_wr<<<blocks(RPAD * DD), TPB, 0, stream>>>(Wr, WrP);
  launch_gemm(stream, priorP, WrP, br, Rg, BE * LL, RPAD, DD, 0);

  for (int it = 0; it < 2; ++it) {
    k_repeat2<<<blocks(NPFc * BE * DD), TPB, 0, stream>>>(pf, extg);
    run_layer(stream, GL, g, LL, BE, extg, NPFc, Rg, maskA, nullptr,
              qb, kb, vb, ab, ob, f1b, f2b);
    k_scale_g<<<blocks(LL * BE * DD), TPB, 0, stream>>>(g, gmask);
    if (it == 1) break;
    k_addio<<<blocks(LL * BE * DD), TPB, 0, stream>>>(g, in_emb, out_emb);
    k_build_extp<<<blocks(EXTP * BP * DD), TPB, 0, stream>>>(g, tf, task_emb, extp);
    run_layer(stream, PL, pf, NPFc, BP, extp, EXTP, nullptr, nullptr, maskE,
              qb, kb, vb, ab, ob, f1b, f2b);
    k_ext2<<<blocks(PP * NPFc * BB * DD), TPB, 0, stream>>>(pf, pair_emb, ext2);
    run_layer(stream, TL, tf, NTFc, BB, ext2, PP * NPFc, nullptr, nullptr, nullptr,
              qb, kb, vb, ab, ob, f1b, f2b);
  }

  // outputs: out (L*BE*11), pf (64*8*256), tf (128*2*256), concatenated
  float* dout = (float*)d_out;
  k_out_proj<<<blocks(LL * BE * OUTD), TPB, 0, stream>>>(g, W_out, b_out, dout);
  const size_t o1 = (size_t)LL * BE * OUTD;
  const size_t o2 = o1 + (size_t)NPFc * BP * DD;
  hipMemcpyAsync(dout + o1, pf, (size_t)NPFc * BP * DD * sizeof(float),
                 hipMemcpyDeviceToDevice, stream);
  hipMemcpyAsync(dout + o2, tf, (size_t)NTFc * BB * DD * sizeof(float),
                 hipMemcpyDeviceToDevice, stream);
}